// DepthDC_30571577212982
// MI455X (gfx1250) — compile-verified
//
#include <hip/hip_runtime.h>

typedef float v2f __attribute__((ext_vector_type(2)));
typedef float v8f __attribute__((ext_vector_type(8)));

#define C_CH   64
#define HDIM   256
#define WDIM   256
#define TILE   16
#define HALO   18                        // TILE + 2 (conv pad 1)
#define HH     (HALO * HALO)             // 324 halo pixels
#define KTOT   576                       // 9 taps * 64 cin (tap-major in LDS)
#define WSTRIDE 580                      // weight row pad: 580 % 64 == 4 -> conflict-free
#define ISTRIDE 68                       // inter row pad:  68 % 64 == 4 -> conflict-free
#define WFLOATS (C_CH * WSTRIDE)         // 37120 floats
#define INTER_FLOATS (HH * ISTRIDE)      // 22032 floats
#define LDS_FLOATS (WFLOATS + INTER_FLOATS + C_CH)   // 59216 floats = 236864 B
#define RSPAN  (HALO * ISTRIDE)          // B row stride in floats (1224)

#define WMMA_F32(a, b, c) __builtin_amdgcn_wmma_f32_16x16x4_f32( \
        false, (a), false, (b), (short)0, (c), false, false)

// sched_group_barrier masks: 0x008 = MFMA/WMMA, 0x100 = DS read
#define SCHED_WMMA(n)  __builtin_amdgcn_sched_group_barrier(0x008, (n), 0)
#define SCHED_DSRD(n)  __builtin_amdgcn_sched_group_barrier(0x100, (n), 0)

__global__ __launch_bounds__(256, 2)     // LDS caps us at 2 waves/SIMD: free the VGPRs
void depthdc_fused_wmma(const float* __restrict__ x,
                        const float* __restrict__ y,
                        const float* __restrict__ fw,
                        const float* __restrict__ fb,
                        float* __restrict__ out) {
    extern __shared__ float smem[];
    float* w_s     = smem;                       // [64][580]  K tap-major: tap*64+cin
    float* inter_s = smem + WFLOATS;             // [324][68]  halo pixel x channel
    float* bias_s  = inter_s + INTER_FLOATS;     // [64]

    const int tid = threadIdx.x;
    const int tx = blockIdx.x, ty = blockIdx.y, n = blockIdx.z;
    const int h0 = ty * TILE, w0 = tx * TILE;

    // ---- cache fuse weights transposed to [cout][tap*64+cin] + bias ----
    for (int idx = tid; idx < C_CH * KTOT; idx += 256) {   // coalesced global reads
        const int cout = idx / KTOT;
        const int rem  = idx - cout * KTOT;
        const int cin  = rem / 9;
        const int tap  = rem - cin * 9;
        w_s[cout * WSTRIDE + tap * C_CH + cin] = fw[idx];
    }
    if (tid < C_CH) bias_s[tid] = fb[tid];

    // ---- stage 1: dynamic depthwise (dilation 2) into LDS, channel-innermost ----
    const size_t xbase = (size_t)n * C_CH * HDIM * WDIM;
    const size_t ybase = (size_t)n * C_CH * 9 * HDIM * WDIM;
    for (int e = tid; e < C_CH * HH; e += 256) {           // 81 iterations, exact
        const int c  = e / HH;                             // coalesced y/x reads
        const int sp = e - c * HH;
        const int ph = sp / HALO, pw = sp - (sp / HALO) * HALO;
        const int gh = h0 - 1 + ph, gw = w0 - 1 + pw;
        float acc = 0.f;
        if ((unsigned)gh < HDIM && (unsigned)gw < WDIM) {
            const float* yp = y + ybase + (((size_t)c * 9) * HDIM + gh) * WDIM + gw;
            const float* xp = x + xbase + (size_t)c * HDIM * WDIM;
            #pragma unroll
            for (int i = 0; i < 3; ++i) {
                const int xh = gh + (i - 1) * 2;
                const bool hok = (unsigned)xh < HDIM;
                #pragma unroll
                for (int j = 0; j < 3; ++j) {
                    const int xw = gw + (j - 1) * 2;
                    const float xv = (hok && (unsigned)xw < WDIM)
                                       ? xp[xh * WDIM + xw] : 0.f;
                    const float yv = yp[(size_t)(i * 3 + j) * HDIM * WDIM];
                    acc = fmaf(xv, yv, acc);
                }
            }
        }
        inter_s[sp * ISTRIDE + c] = acc;
    }
    __syncthreads();

    // ---- stage 2: 3x3 conv as 64x576 GEMM via v_wmma_f32_16x16x4_f32 ----
    // Wave block: 2 M-tiles x 4 output rows; K tap-major so each lane's two K
    // values are adjacent cin -> single ds_load_b64. Double-buffered pipeline
    // over K-pairs; sched_group_barriers interleave [4 WMMA][3 DS-read] so
    // prefetch loads issue ~12-16 WMMAs ahead of their consumers.
    const int lane = tid & 31;
    const int wv   = tid >> 5;
    const int mg   = wv & 1;          // cout half: couts mg*32 .. mg*32+31
    const int rg   = wv >> 1;         // rows rg*4 .. rg*4+3
    const int col  = lane & 15;       // N index (pixel col) & A row (M)
    const int hi   = lane >> 4;       // lane half -> K slots {0,1} vs {2,3}
    const int ksub = hi * 2;
    const int rowoff = rg * 4;

    v8f acc0[4] = {}, acc1[4] = {};   // two 16-cout tiles x 4 rows

    const float* arow0 = w_s + (mg * 32 + col) * WSTRIDE + ksub;
    const float* arow1 = arow0 + 16 * WSTRIDE;

    for (int tap = 0; tap < 9; ++tap) {
        const int ti = tap / 3;
        const int tj = tap - ti * 3;
        const float* ap0 = arow0 + tap * C_CH;
        const float* ap1 = arow1 + tap * C_CH;
        const float* bp  = inter_s + ((rowoff + ti) * HALO + tj + col) * ISTRIDE + ksub;

        v2f a0[2][2], a1[2][2];       // [buf][k-sub]
        v2f b [2][2][4];              // [buf][k-sub][row]

        // prime buffer 0 with K-pair kc = {0,1}
        a0[0][0] = *(const v2f*)(ap0);      a0[0][1] = *(const v2f*)(ap0 + 4);
        a1[0][0] = *(const v2f*)(ap1);      a1[0][1] = *(const v2f*)(ap1 + 4);
        #pragma unroll
        for (int r = 0; r < 4; ++r) {
            b[0][0][r] = *(const v2f*)(bp     + r * RSPAN);
            b[0][1][r] = *(const v2f*)(bp + 4 + r * RSPAN);
        }

        #pragma unroll
        for (int kcp = 0; kcp < 8; ++kcp) {          // 8 K-pairs (16 k-steps)
            const int cur = kcp & 1, nxt = cur ^ 1;
            if (kcp < 7) {                           // prefetch next K-pair
                const int kf = (kcp + 1) * 8;        // float offset of next pair
                a0[nxt][0] = *(const v2f*)(ap0 + kf);
                a0[nxt][1] = *(const v2f*)(ap0 + kf + 4);
                a1[nxt][0] = *(const v2f*)(ap1 + kf);
                a1[nxt][1] = *(const v2f*)(ap1 + kf + 4);
                #pragma unroll
                for (int r = 0; r < 4; ++r) {
                    b[nxt][0][r] = *(const v2f*)(bp + kf     + r * RSPAN);
                    b[nxt][1][r] = *(const v2f*)(bp + kf + 4 + r * RSPAN);
                }
            }
            #pragma unroll
            for (int s = 0; s < 2; ++s) {            // 16 WMMAs on current buffer
                #pragma unroll
                for (int r = 0; r < 4; ++r) {
                    acc0[r] = WMMA_F32(a0[cur][s], b[cur][s][r], acc0[r]);
                    acc1[r] = WMMA_F32(a1[cur][s], b[cur][s][r], acc1[r]);
                }
            }
            // schedule shape: spread the 12 prefetch DS-reads under the 16 WMMAs
            SCHED_WMMA(4); SCHED_DSRD(3);
            SCHED_WMMA(4); SCHED_DSRD(3);
            SCHED_WMMA(4); SCHED_DSRD(3);
            SCHED_WMMA(4); SCHED_DSRD(3);
        }
    }

    // ---- epilogue: bias + LeakyReLU(0.2), coalesced stores ----
    // D layout: lane = N (pixel col), VGPR v -> M = v (lanes 0-15) / v+8 (16-31)
    float biasv0[8], biasv1[8];
    #pragma unroll
    for (int v = 0; v < 8; ++v) {
        biasv0[v] = bias_s[mg * 32      + v + hi * 8];
        biasv1[v] = bias_s[mg * 32 + 16 + v + hi * 8];
    }

    const int gw = w0 + col;
    const size_t nbase = (size_t)n * C_CH * HDIM * WDIM;
    #pragma unroll
    for (int r = 0; r < 4; ++r) {
        const int gh = h0 + rowoff + r;
        const size_t base = nbase + (size_t)gh * WDIM + gw;
        #pragma unroll
        for (int v = 0; v < 8; ++v) {
            const int co0 = mg * 32      + v + hi * 8;
            const int co1 = mg * 32 + 16 + v + hi * 8;
            float v0 = acc0[r][v] + biasv0[v];
            float v1 = acc1[r][v] + biasv1[v];
            v0 = (v0 >= 0.f) ? v0 : 0.2f * v0;
            v1 = (v1 >= 0.f) ? v1 : 0.2f * v1;
            out[base + (size_t)co0 * HDIM * WDIM] = v0;
            out[base + (size_t)co1 * HDIM * WDIM] = v1;
        }
    }
}

extern "C" void kernel_launch(void* const* d_in, const int* in_sizes, int n_in,
                              void* d_out, int out_size, void* d_ws, size_t ws_size,
                              hipStream_t stream) {
    const float* x  = (const float*)d_in[0];
    const float* y  = (const float*)d_in[1];
    const float* fw = (const float*)d_in[2];
    const float* fb = (const float*)d_in[3];
    float* out = (float*)d_out;

    const size_t shmem = LDS_FLOATS * sizeof(float);   // ~231 KB (< 320 KB/WGP)
    hipFuncSetAttribute(reinterpret_cast<const void*>(depthdc_fused_wmma),
                        hipFuncAttributeMaxDynamicSharedMemorySize, (int)shmem);

    dim3 grid(WDIM / TILE, HDIM / TILE, 2);            // 16 x 16 x 2 blocks
    hipLaunchKernelGGL(depthdc_fused_wmma, grid, dim3(256), shmem, stream,
                       x, y, fw, fb, out);
}